// SuperLoss_8727373545535
// MI455X (gfx1250) — compile-verified
//
#include <hip/hip_runtime.h>
#include <math.h>

#define NROWS 65536
#define NCOLS 1000
#define TAU_F 6.907755278982137f    /* log(1000) */
#define LAM_F 2.0f
#define INV_E_F 0.36787944117144233f
#define E_F 2.718281828459045f
#define LOG2E_F 1.4426950408889634f

typedef float v2f __attribute__((ext_vector_type(2)));
typedef float v8f __attribute__((ext_vector_type(8)));

// Fast exp2: single v_exp_f32, no range bookkeeping (args are in [-18, 0] here).
__device__ __forceinline__ float fast_exp2(float x) {
#if __has_builtin(__builtin_amdgcn_exp2f)
    return __builtin_amdgcn_exp2f(x);
#else
    return __expf(x * 0.6931471805599453f);   // fallback: still native v_exp
#endif
}

__device__ __forceinline__ float lambertw20(float y) {
    // exact transcription of the reference _lambertw (float32, 20 Halley iters)
    y = fmaxf(y, -INV_E_F + 1e-7f);
    float p  = sqrtf(fmaxf(2.0f * (E_F * y + 1.0f), 0.0f));
    float wb = -1.0f + p - p * p * (1.0f / 3.0f);
    float w  = (y > 1.0f) ? logf(fmaxf(y, 1e-6f)) : wb;
    #pragma unroll 1
    for (int i = 0; i < 20; ++i) {
        float ew    = expf(w);                 // precise libm exp in the tail
        float f     = w * ew - y;
        float wp1   = w + 1.0f;
        float denom = ew * wp1 - (w + 2.0f) * f / (2.0f * wp1);
        denom       = (fabsf(denom) < 1e-12f) ? 1e-12f : denom;
        w           = w - f / denom;
    }
    return w;
}

// One wave (32 lanes) handles 16 rows. Lane L: row = rb + (L & 15),
// half = L >> 4 selects the column-pair parity so the fp32 A-matrix layout
// of V_WMMA_F32_16X16X4_F32 is produced directly by float2 loads.
__global__ __launch_bounds__(256) void superloss_rows_kernel(
    const float* __restrict__ logits,
    const int*   __restrict__ targets,
    float*       __restrict__ row_loss)
{
    __shared__ float s_se[8 * 16];   // 8 waves/block * 16 rows

    const int lane = threadIdx.x & 31;
    const int wave = threadIdx.x >> 5;
    const int rb   = (blockIdx.x * 8 + wave) * 16;
    const int r    = lane & 15;
    const int half = lane >> 4;

    const float* rowp = logits + (size_t)(rb + r) * NCOLS;

    // ---- pass 1: row max (coalesced float4, 125 per lane, 1000 = 125*8) ----
    float m = -INFINITY;
    const float4* p4 = (const float4*)rowp;
    #pragma unroll 5
    for (int j = 0; j < 125; ++j) {
        float4 v = p4[2 * j + half];
        m = fmaxf(m, fmaxf(fmaxf(v.x, v.y), fmaxf(v.z, v.w)));
    }
    m = fmaxf(m, __shfl_xor(m, 16, 32));   // combine the two half-row scans
    const float mm = m * LOG2E_F;          // exp(x-m) = exp2(fma(x, log2e, -mm))

    // ---- pass 2: sum of exp via V_WMMA_F32_16X16X4_F32, B = ones ----
    // D[i,n] += sum_k A[i,k]  -> every column of D is the fp32 row-sum of exp.
    // Per element: 1 v_fma + 1 v_exp_f32 (trans co-executes with VALU).
    v8f d0 = {0.f,0.f,0.f,0.f,0.f,0.f,0.f,0.f};
    v8f d1 = {0.f,0.f,0.f,0.f,0.f,0.f,0.f,0.f};
    const v2f ones = {1.0f, 1.0f};
    const float2* p2 = (const float2*)rowp;
    #pragma unroll 2
    for (int j = 0; j < 250; j += 2) {
        float2 va = p2[2 * j + half];                  // cols 4j + 2*half (+0,+1)
        float2 vb = p2[2 * (j + 1) + half];
        v2f a0, a1;
        a0.x = fast_exp2(__builtin_fmaf(va.x, LOG2E_F, -mm));
        a0.y = fast_exp2(__builtin_fmaf(va.y, LOG2E_F, -mm));
        a1.x = fast_exp2(__builtin_fmaf(vb.x, LOG2E_F, -mm));
        a1.y = fast_exp2(__builtin_fmaf(vb.y, LOG2E_F, -mm));
        d0 = __builtin_amdgcn_wmma_f32_16x16x4_f32(false, a0, false, ones,
                                                   (short)0, d0, false, false);
        d1 = __builtin_amdgcn_wmma_f32_16x16x4_f32(false, a1, false, ones,
                                                   (short)0, d1, false, false);
    }

    // D layout: VGPR k, lanes 0-15 -> row k (N=lane); lanes 16-31 -> row k+8.
    // Lane 0 holds rows rb..rb+7 in d[0..7]; lane 16 holds rows rb+8..rb+15.
    if ((lane & 15) == 0) {
        #pragma unroll
        for (int k = 0; k < 8; ++k)
            s_se[wave * 16 + half * 8 + k] = d0[k] + d1[k];
    }
    __syncthreads();

    // ---- per-row tail: ce, Lambert W, SuperLoss ----
    if (lane < 16) {
        const int   row    = rb + lane;
        const float sumexp = s_se[wave * 16 + lane];
        const int   tgt    = targets[row];
        const float lt     = logits[(size_t)row * NCOLS + tgt];

        const float ce = m + logf(sumexp) - lt;          // m = max of row rb+lane
        const float y  = 0.5f * fmaxf(-2.0f * INV_E_F, (ce - TAU_F) / LAM_F);
        const float w  = lambertw20(y);
        const float sigma = expf(-w);                    // log(sigma) == -w
        row_loss[row] = (ce - TAU_F) * sigma + LAM_F * w * w;
    }
}

// Deterministic fixed-order reduction of the 65536 per-row losses.
__global__ __launch_bounds__(1024) void superloss_reduce_kernel(
    const float* __restrict__ x, float* __restrict__ out)
{
    __shared__ float s[1024];
    const int t = threadIdx.x;
    float acc = 0.0f;
    #pragma unroll 4
    for (int j = 0; j < NROWS / 1024; ++j)
        acc += x[t + 1024 * j];
    s[t] = acc;
    __syncthreads();
    #pragma unroll
    for (int off = 512; off > 0; off >>= 1) {
        if (t < off) s[t] += s[t + off];
        __syncthreads();
    }
    if (t == 0) out[0] = s[0] * (1.0f / (float)NROWS);
}

extern "C" void kernel_launch(void* const* d_in, const int* in_sizes, int n_in,
                              void* d_out, int out_size, void* d_ws, size_t ws_size,
                              hipStream_t stream) {
    (void)in_sizes; (void)n_in; (void)out_size; (void)ws_size;
    const float* logits  = (const float*)d_in[0];
    const int*   targets = (const int*)d_in[1];
    float* row_loss = (float*)d_ws;          // 65536 floats = 256 KB scratch
    float* out      = (float*)d_out;

    // 16 rows per wave, 8 waves (256 thr) per block -> 128 rows/block -> 512 blocks
    superloss_rows_kernel<<<NROWS / 128, 256, 0, stream>>>(logits, targets, row_loss);
    superloss_reduce_kernel<<<1, 1024, 0, stream>>>(row_loss, out);
}